// MultiChannelSequentialCircuit_84902913507874
// MI455X (gfx1250) — compile-verified
//
#include <hip/hip_runtime.h>

// CDNA5 / gfx1250, wave32. Quantum state-vector sim via fp32 WMMA group-GEMMs.

typedef float v2f __attribute__((ext_vector_type(2)));
typedef float v8f __attribute__((ext_vector_type(8)));

#if defined(__HIP_DEVICE_COMPILE__) && \
    !__has_builtin(__builtin_amdgcn_wmma_f32_16x16x4_f32)
#error "device pass: missing __builtin_amdgcn_wmma_f32_16x16x4_f32"
#endif

#define NQ      16
#define NSTATES 65536
#define CIN     3
#define BATCH   1024
#define TPB     256

__device__ __forceinline__ v8f wmma_f32(v2f a, v2f b, v8f c) {
#if defined(__HIP_DEVICE_COMPILE__)
  // D = A(16x4) x B(4x16) + C(16x16), fp32
  return __builtin_amdgcn_wmma_f32_16x16x4_f32(false, a, false, b, (short)0, c,
                                               false, false);
#else
  (void)a; (void)b;
  return c;  // host pass: never executed
#endif
}

// Inverse of the CX-ring permutation L (ring: CX(q,q+1) q=0..14, then CX(15,0)).
// Forward: y_k = x_0^..^x_k (k>=1), y_0 = x_1^..^x_15.
// Inverse: x_0 = y_0^y_15, x_1 = y_0^y_1^y_15, x_k = y_{k-1}^y_k (k>=2).
__device__ __forceinline__ unsigned ring_linv(unsigned y) {
  unsigned x = (y ^ (y << 1)) & 0xFFFFu;
  if (y & 0x8000u) x ^= 0x3u;
  return x;
}

// complex 2x2 row-major: {r00,i00, r01,i01, r10,i10, r11,i11}; C = A*B
__device__ __forceinline__ void cmul2x2(const float* A, const float* B, float* C) {
#pragma unroll
  for (int i = 0; i < 2; ++i)
#pragma unroll
    for (int j = 0; j < 2; ++j) {
      float re = 0.f, im = 0.f;
#pragma unroll
      for (int k = 0; k < 2; ++k) {
        float ar = A[(i * 2 + k) * 2], ai = A[(i * 2 + k) * 2 + 1];
        float br = B[(k * 2 + j) * 2], bi = B[(k * 2 + j) * 2 + 1];
        re += ar * br - ai * bi;
        im += ar * bi + ai * br;
      }
      C[(i * 2 + j) * 2] = re;
      C[(i * 2 + j) * 2 + 1] = im;
    }
}

// Apply one 4-qubit group's 16x16 complex gate matrix M to the LDS tile,
// in place, via V_WMMA_F32_16X16X4_F32.
// Tile has 4096 complex elements; element (m, col j) lives at
//   addr(m,j) = ((j & ~(gs-1)) << 4) + m*gs + (j & (gs-1)),  gs = 16^g.
// 256 columns -> 16 blocks of N=16; 8 waves handle 2 blocks each.
__device__ __forceinline__ void apply_group(int tid, int gs,
                                            const float* __restrict__ Mr,
                                            const float* __restrict__ Mi,
                                            float* XR, float* XI) {
  const int lane = tid & 31;
  const int wave = tid >> 5;
  const int half = lane >> 4;   // 0: lanes 0-15, 1: lanes 16-31
  const int ln   = lane & 15;
#pragma unroll
  for (int bb = 0; bb < 2; ++bb) {
    const int blk  = wave * 2 + bb;
    const int j    = blk * 16 + ln;
    const int base = ((j & ~(gs - 1)) << 4) + (j & (gs - 1));

    v2f ar[4], ai[4], an[4], br[4], bi[4];
#pragma unroll
    for (int k = 0; k < 4; ++k) {
      const int c0 = 4 * k + 2 * half;  // K index per ISA f32 A layout
      ar[k].x = Mr[ln * 16 + c0];       ar[k].y = Mr[ln * 16 + c0 + 1];
      ai[k].x = Mi[ln * 16 + c0];       ai[k].y = Mi[ln * 16 + c0 + 1];
      an[k].x = -ai[k].x;               an[k].y = -ai[k].y;
      br[k].x = XR[base + c0 * gs];     br[k].y = XR[base + (c0 + 1) * gs];
      bi[k].x = XI[base + c0 * gs];     bi[k].y = XI[base + (c0 + 1) * gs];
    }
    v8f accR = {0.f, 0.f, 0.f, 0.f, 0.f, 0.f, 0.f, 0.f};
    v8f accI = {0.f, 0.f, 0.f, 0.f, 0.f, 0.f, 0.f, 0.f};
#pragma unroll
    for (int k = 0; k < 4; ++k) {
      accR = wmma_f32(ar[k], br[k], accR);   // + Mr*Xr
      accR = wmma_f32(an[k], bi[k], accR);   // - Mi*Xi
      accI = wmma_f32(ar[k], bi[k], accI);   // + Mr*Xi
      accI = wmma_f32(ai[k], br[k], accI);   // + Mi*Xr
    }
#pragma unroll
    for (int r = 0; r < 8; ++r) {
      const int mp = r + 8 * half;          // D row per ISA f32 C/D layout
      XR[base + mp * gs] = accR[r];
      XI[base + mp * gs] = accI[r];
    }
  }
}

__global__ __launch_bounds__(TPB) void qsim_kernel(
    const float* __restrict__ weights,  // (3,2,16,3)
    const float* __restrict__ patch,    // (1024,3,16)
    float* __restrict__ out,            // (1024,16)
    float* __restrict__ ws, int nwg) {
  __shared__ float sRe[4096];
  __shared__ float sIm[4096];
  __shared__ float sM[4][2][256];   // [group][re/im][m'*16+m]
  __shared__ float sGates[2][16][8];// [layer V/W][qubit][2x2 complex]
  __shared__ float sV[16][4];       // per-qubit product vector (b=0 re,im, b=1 re,im)
  __shared__ float sT[4][16][2];    // nibble tables for product state

  const int tid  = threadIdx.x;
  const int wg   = blockIdx.x;
  float* gR = ws + (size_t)wg * (size_t)(2 * NSTATES);
  float* gI = gR + NSTATES;

  for (int b = wg; b < BATCH; b += nwg) {
    for (int ch = 0; ch < CIN; ++ch) {
      // ---- build per-qubit gates: V = Urot0 * RY(patch), W = Urot1 ----
      if (tid < NQ) {
        const int q = tid;
        float th = patch[(size_t)b * (CIN * NQ) + ch * NQ + q];
        float ce = cosf(0.5f * th), se = sinf(0.5f * th);
        float Renc[8] = {ce, 0.f, -se, 0.f, se, 0.f, ce, 0.f};
#pragma unroll
        for (int l = 0; l < 2; ++l) {
          const float* wp = weights + ch * (2 * NQ * 3) + l * (NQ * 3) + q * 3;
          float a = wp[0], bb = wp[1], c = wp[2];
          float ca = cosf(0.5f * a),  sa = sinf(0.5f * a);
          float cb = cosf(0.5f * bb), sb = sinf(0.5f * bb);
          float cz = cosf(0.5f * c),  sz = sinf(0.5f * c);
          float RX[8]  = {ca, 0.f, 0.f, -sa, 0.f, -sa, ca, 0.f};
          float RYm[8] = {cb, 0.f, -sb, 0.f, sb, 0.f, cb, 0.f};
          float RZm[8] = {cz, -sz, 0.f, 0.f, 0.f, 0.f, cz, sz};
          float T1[8], U[8];
          cmul2x2(RYm, RX, T1);
          cmul2x2(RZm, T1, U);
          if (l == 0) {
            float V[8];
            cmul2x2(U, Renc, V);
#pragma unroll
            for (int k = 0; k < 8; ++k) sGates[0][q][k] = V[k];
          } else {
#pragma unroll
            for (int k = 0; k < 8; ++k) sGates[1][q][k] = U[k];
          }
        }
      }
      __syncthreads();

      if (ch == 0) {
        // product-state vectors v_q = V_q * (1/sqrt2, 1/sqrt2)
        if (tid < NQ) {
          const float s = 0.70710678118654752f;
#pragma unroll
          for (int bp = 0; bp < 2; ++bp) {
            float vr = s * (sGates[0][tid][(bp * 2) * 2]     + sGates[0][tid][(bp * 2 + 1) * 2]);
            float vi = s * (sGates[0][tid][(bp * 2) * 2 + 1] + sGates[0][tid][(bp * 2 + 1) * 2 + 1]);
            sV[tid][bp * 2] = vr;
            sV[tid][bp * 2 + 1] = vi;
          }
        }
        __syncthreads();
        if (tid < 64) {  // nibble tables T_g[m] = prod_j v_{4g+j}[m_j]
          int g = tid >> 4, m = tid & 15;
          float pr = 1.f, pi = 0.f;
#pragma unroll
          for (int j = 0; j < 4; ++j) {
            int q = 4 * g + j, bit = (m >> j) & 1;
            float vr = sV[q][bit * 2], vi = sV[q][bit * 2 + 1];
            float nr = pr * vr - pi * vi, ni = pr * vi + pi * vr;
            pr = nr; pi = ni;
          }
          sT[g][m][0] = pr;
          sT[g][m][1] = pi;
        }
        __syncthreads();
      }

      // ch==0: V-layer folded into the product-state generator (lay starts at 1)
      for (int lay = (ch == 0) ? 1 : 0; lay < 2; ++lay) {
        // ---- build 16x16 group matrices M_g = tensor of 4 qubit gates ----
        for (int idx = tid; idx < 1024; idx += TPB) {
          int g = idx >> 8, e = idx & 255, mp = e >> 4, mm = e & 15;
          float pr = 1.f, pi = 0.f;
#pragma unroll
          for (int j = 0; j < 4; ++j) {
            int q = 4 * g + j;
            int bi_ = (mp >> j) & 1, bj = (mm >> j) & 1;
            float grr = sGates[lay][q][(bi_ * 2 + bj) * 2];
            float gii = sGates[lay][q][(bi_ * 2 + bj) * 2 + 1];
            float nr = pr * grr - pi * gii;
            float ni = pr * gii + pi * grr;
            pr = nr; pi = ni;
          }
          sM[g][0][e] = pr;
          sM[g][1][e] = pi;
        }
        __syncthreads();

        const bool gen = (ch == 0 && lay == 1);  // generate product state on the fly

        // ---- LOW PASS: groups 0..2 on contiguous 4096-element tiles ----
        // Read through the pending ring permutation (always pending here).
        for (int t = 0; t < 16; ++t) {
          for (int idx = tid; idx < 4096; idx += TPB) {
            unsigned y = (unsigned)(t * 4096 + idx);
            unsigned x = ring_linv(y);
            if (gen) {
              float pr = sT[0][x & 15][0], pi = sT[0][x & 15][1];
#pragma unroll
              for (int g = 1; g < 4; ++g) {
                int m = (x >> (4 * g)) & 15;
                float tr = sT[g][m][0], ti = sT[g][m][1];
                float nr = pr * tr - pi * ti, ni = pr * ti + pi * tr;
                pr = nr; pi = ni;
              }
              sRe[idx] = pr;
              sIm[idx] = pi;
            } else {
              sRe[idx] = gR[x];
              sIm[idx] = gI[x];
            }
          }
          __syncthreads();
          for (int g = 0; g < 3; ++g) {
            apply_group(tid, 1 << (4 * g), &sM[g][0][0], &sM[g][1][0], sRe, sIm);
            __syncthreads();
          }
          for (int idx = tid; idx < 4096; idx += TPB) {
            gR[t * 4096 + idx] = sRe[idx];
            gI[t * 4096 + idx] = sIm[idx];
          }
          __syncthreads();
        }

        // ---- HIGH PASS: group 3 (bits 12..15), plain strided tiles ----
        for (int t = 0; t < 16; ++t) {
          for (int idx = tid; idx < 4096; idx += TPB) {
            int m = idx >> 8, r = idx & 255;
            unsigned y = (unsigned)(m * 4096 + t * 256 + r);
            sRe[idx] = gR[y];
            sIm[idx] = gI[y];
          }
          __syncthreads();
          apply_group(tid, 256, &sM[3][0][0], &sM[3][1][0], sRe, sIm);
          __syncthreads();
          for (int idx = tid; idx < 4096; idx += TPB) {
            int m = idx >> 8, r = idx & 255;
            unsigned y = (unsigned)(m * 4096 + t * 256 + r);
            gR[y] = sRe[idx];
            gI[y] = sIm[idx];
          }
          __syncthreads();
        }
      }  // layer
    }    // channel

    // ---- measurement: pending ring folded into parity masks ----
    // (Lx)_0 = parity(x & 0xFFFE); (Lx)_q = parity(x & (2^{q+1}-1)), q>=1
    float acc[16];
#pragma unroll
    for (int q = 0; q < 16; ++q) acc[q] = 0.f;
    for (int idx = tid; idx < NSTATES; idx += TPB) {
      float re = gR[idx], im = gI[idx];
      float p = re * re + im * im;
      unsigned x = (unsigned)idx;
#pragma unroll
      for (int q = 0; q < 16; ++q) {
        unsigned mask = q ? ((2u << q) - 1u) : 0xFFFEu;
        acc[q] += (__popc(x & mask) & 1) ? -p : p;
      }
    }
    __syncthreads();
#pragma unroll
    for (int q = 0; q < 16; ++q) sRe[q * 256 + tid] = acc[q];
    __syncthreads();
    if (tid < 16) {
      float s = 0.f;
      for (int t2 = 0; t2 < 256; ++t2) s += sRe[tid * 256 + t2];
      out[(size_t)b * 16 + tid] = s;
    }
    __syncthreads();
  }  // batch loop
}

extern "C" void kernel_launch(void* const* d_in, const int* in_sizes, int n_in,
                              void* d_out, int out_size, void* d_ws, size_t ws_size,
                              hipStream_t stream) {
  (void)in_sizes; (void)n_in; (void)out_size;
  const float* weights = (const float*)d_in[0];
  const float* patch   = (const float*)d_in[1];
  float* out = (float*)d_out;

  // One 512KB state (re+im fp32) per resident workgroup; size grid from ws.
  const size_t perWG = (size_t)(2 * NSTATES) * sizeof(float);
  size_t n = ws_size / perWG;
  if (n < 1) n = 1;
  if (n > BATCH) n = BATCH;
  const int nwg = (int)n;

  qsim_kernel<<<dim3(nwg), dim3(TPB), 0, stream>>>(weights, patch, out,
                                                   (float*)d_ws, nwg);
}